// ContinuousExpansionLayer_2568390443335
// MI455X (gfx1250) — compile-verified
//
#include <hip/hip_runtime.h>
#include <hip/hip_bf16.h>
#include <math.h>

#define B_SZ   32
#define S_LEN  4096
#define T_LEN  8192
#define DIN    128
#define DT     256
#define LDSTR  260   // 256 + 4 pad floats: row stride ≡ 4 mod 64 banks -> conflict-free WMMA A reads

typedef __attribute__((ext_vector_type(2))) float v2f;
typedef __attribute__((ext_vector_type(8))) float v8f;

__device__ __forceinline__ float gelu_exact(float x) {
    return 0.5f * x * (1.0f + erff(x * 0.70710678118654752f));
}

// ---------------------------------------------------------------------------
// Kernel 1: s_lens[b] = sum(s_mask[b,:]); softmax(neighbor_weights) -> ws[32..34]
// ---------------------------------------------------------------------------
__global__ __launch_bounds__(256) void prep_kernel(const float* __restrict__ smask,
                                                   const float* __restrict__ nw,
                                                   float* __restrict__ ws) {
    __shared__ float red[256];
    const int b = blockIdx.x;
    float s = 0.0f;
    for (int i = threadIdx.x; i < S_LEN; i += 256) s += smask[(size_t)b * S_LEN + i];
    red[threadIdx.x] = s;
    __syncthreads();
    for (int off = 128; off > 0; off >>= 1) {
        if ((int)threadIdx.x < off) red[threadIdx.x] += red[threadIdx.x + off];
        __syncthreads();
    }
    if (threadIdx.x == 0) ws[b] = red[0];
    if (b == 0 && threadIdx.x == 0) {
        float m  = fmaxf(nw[0], fmaxf(nw[1], nw[2]));
        float e0 = expf(nw[0] - m), e1 = expf(nw[1] - m), e2 = expf(nw[2] - m);
        float inv = 1.0f / (e0 + e1 + e2);
        ws[32] = e0 * inv; ws[33] = e1 * inv; ws[34] = e2 * inv;
    }
}

// ---------------------------------------------------------------------------
// Kernel 2: pos_emb[t][d] (batch-independent) into scratch. One block per t.
// ---------------------------------------------------------------------------
__global__ __launch_bounds__(128) void posemb_kernel(const float* __restrict__ pew1,
                                                     const float* __restrict__ peb1,
                                                     const float* __restrict__ pew2,
                                                     const float* __restrict__ peb2,
                                                     float* __restrict__ pos_out) {
    __shared__ float h[64];
    const int t = blockIdx.x;
    const float pos = (float)t * (1.0f / (float)(T_LEN - 1));
    if (threadIdx.x < 64) {
        float x = pos * pew1[threadIdx.x] + peb1[threadIdx.x];
        h[threadIdx.x] = gelu_exact(x);
    }
    __syncthreads();
    const int d = threadIdx.x;
    float acc = peb2[d];
#pragma unroll 8
    for (int j = 0; j < 64; ++j) acc += h[j] * pew2[j * DIN + d];
    pos_out[(size_t)t * DIN + d] = acc;
}

// ---------------------------------------------------------------------------
// Main fused kernel: gather+blend -> combined in LDS -> WMMA f32 GEMM with
// pt_w -> bias+GELU -> LayerNorm -> t_mask -> store. 32 t-rows per block.
// ---------------------------------------------------------------------------
__global__ __launch_bounds__(256) void cel_main_kernel(const float* __restrict__ emb,
                                                       const float* __restrict__ smask,
                                                       const float* __restrict__ tmask,
                                                       const float* __restrict__ ptw,
                                                       const float* __restrict__ ptb,
                                                       const float* __restrict__ lng,
                                                       const float* __restrict__ lnb,
                                                       const float* __restrict__ ws,
                                                       float* __restrict__ out) {
    __shared__ float lds[32 * LDSTR];

    const int wave = threadIdx.x >> 5;
    const int lane = threadIdx.x & 31;
    const int tiles_per_b = T_LEN / 32;
    const int b  = blockIdx.x / tiles_per_b;
    const int t0 = (blockIdx.x % tiles_per_b) * 32;

    const float slen = ws[b];
    const float w0 = ws[32], w1 = ws[33], w2 = ws[34];
    const float* __restrict__ posemb = ws + 64;

    // ---- Stage combined[32][256] into LDS (cols 0..127 blended, 128..255 pos_emb)
#pragma unroll
    for (int rr = 0; rr < 4; ++rr) {
        const int r = wave * 4 + rr;
        const int t = t0 + r;
        const float pos = (float)t * (1.0f / (float)(T_LEN - 1));
        const float sp  = pos * (slen - 1.0f);
        int cur = (int)sp;
        int prv = cur - 1; if (prv < 0) prv = 0;
        int nxt = cur + 1; if (nxt > S_LEN - 1) nxt = S_LEN - 1;
        const size_t base = (size_t)b * S_LEN;
        const float4* ep = (const float4*)(emb + (base + prv) * DIN);
        const float4* ec = (const float4*)(emb + (base + cur) * DIN);
        const float4* en = (const float4*)(emb + (base + nxt) * DIN);
        const float mp = w0 * smask[base + prv];
        const float mc = w1 * smask[base + cur];
        const float mn = w2 * smask[base + nxt];
        const float4 vp = ep[lane], vc = ec[lane], vn = en[lane];
        float4 o;
        o.x = mp * vp.x + mc * vc.x + mn * vn.x;
        o.y = mp * vp.y + mc * vc.y + mn * vn.y;
        o.z = mp * vp.z + mc * vc.z + mn * vn.z;
        o.w = mp * vp.w + mc * vc.w + mn * vn.w;
        *(float4*)&lds[r * LDSTR + lane * 4] = o;
        const float4 pe = ((const float4*)(posemb + (size_t)t * DIN))[lane];
        *(float4*)&lds[r * LDSTR + DIN + lane * 4] = pe;
    }
    __syncthreads();

    // ---- GEMM: wave handles 2 M-subtiles x 2 N-tiles (32x32 patch), K=256 in steps of 4
    const int n0   = wave * 32;
    const int koff = (lane >> 4) * 2;   // WMMA f32 A/B: lanes 0-15 hold K={0,1}, 16-31 K={2,3}
    const int mr   = lane & 15;
    v8f c00 = {}, c01 = {}, c10 = {}, c11 = {};
    for (int k0 = 0; k0 < DT; k0 += 4) {
        const v2f a0 = *(const v2f*)&lds[mr * LDSTR + k0 + koff];
        const v2f a1 = *(const v2f*)&lds[(16 + mr) * LDSTR + k0 + koff];
        const float* bp = ptw + (size_t)(k0 + koff) * DT + (lane & 15);
        v2f b0, b1;
        b0.x = bp[n0];            b0.y = bp[n0 + DT];
        b1.x = bp[n0 + 16];       b1.y = bp[n0 + 16 + DT];
        c00 = __builtin_amdgcn_wmma_f32_16x16x4_f32(false, a0, false, b0, (short)0, c00, false, false);
        c01 = __builtin_amdgcn_wmma_f32_16x16x4_f32(false, a0, false, b1, (short)0, c01, false, false);
        c10 = __builtin_amdgcn_wmma_f32_16x16x4_f32(false, a1, false, b0, (short)0, c10, false, false);
        c11 = __builtin_amdgcn_wmma_f32_16x16x4_f32(false, a1, false, b1, (short)0, c11, false, false);
    }
    __syncthreads();   // all waves done reading A tile before LDS reuse

    // ---- bias + exact GELU, write activations back into the (reused) LDS tile
    {
        const int nA = n0 + (lane & 15);
        const int nB = nA + 16;
        const int mb = 8 * (lane >> 4);   // C/D layout: VGPR j = rows j (lanes 0-15), j+8 (16-31)
        const float bA = ptb[nA], bB = ptb[nB];
#pragma unroll
        for (int j = 0; j < 8; ++j) {
            const int mA = mb + j;
            lds[mA * LDSTR + nA]        = gelu_exact(c00[j] + bA);
            lds[mA * LDSTR + nB]        = gelu_exact(c01[j] + bB);
            lds[(16 + mA) * LDSTR + nA] = gelu_exact(c10[j] + bA);
            lds[(16 + mA) * LDSTR + nB] = gelu_exact(c11[j] + bB);
        }
    }
    __syncthreads();

    // ---- LayerNorm over 256, t_mask, coalesced store. Each wave: 4 rows.
#pragma unroll
    for (int rr = 0; rr < 4; ++rr) {
        const int r = wave * 4 + rr;
        float s = 0.0f, ss = 0.0f;
#pragma unroll
        for (int i = 0; i < 8; ++i) {
            const float x = lds[r * LDSTR + lane + 32 * i];
            s += x; ss += x * x;
        }
#pragma unroll
        for (int off = 16; off > 0; off >>= 1) {
            s  += __shfl_xor(s,  off, 32);
            ss += __shfl_xor(ss, off, 32);
        }
        const float mu  = s  * (1.0f / (float)DT);
        const float var = ss * (1.0f / (float)DT) - mu * mu;
        const float rs  = rsqrtf(var + 1e-5f);
        const int   t   = t0 + r;
        const float tm  = tmask[(size_t)b * T_LEN + t];
        float* op = out + ((size_t)b * T_LEN + t) * DT;
#pragma unroll
        for (int i = 0; i < 8; ++i) {
            const int n = lane + 32 * i;
            const float x = lds[r * LDSTR + n];
            op[n] = ((x - mu) * rs * lng[n] + lnb[n]) * tm;
        }
    }
}

// ---------------------------------------------------------------------------
extern "C" void kernel_launch(void* const* d_in, const int* in_sizes, int n_in,
                              void* d_out, int out_size, void* d_ws, size_t ws_size,
                              hipStream_t stream) {
    const float* emb   = (const float*)d_in[0];   // [B,S,DIN]
    const float* smask = (const float*)d_in[1];   // [B,S]
    const float* tmask = (const float*)d_in[2];   // [B,T]
    // d_in[3] = target_length (== T_LEN), unused
    const float* pew1  = (const float*)d_in[4];   // [1,64]
    const float* peb1  = (const float*)d_in[5];   // [64]
    const float* pew2  = (const float*)d_in[6];   // [64,128]
    const float* peb2  = (const float*)d_in[7];   // [128]
    const float* ptw   = (const float*)d_in[8];   // [256,256]
    const float* ptb   = (const float*)d_in[9];   // [256]
    const float* lng   = (const float*)d_in[10];  // [256]
    const float* lnb   = (const float*)d_in[11];  // [256]
    const float* nw    = (const float*)d_in[12];  // [3]

    float* ws = (float*)d_ws;
    // ws layout (floats): [0..31]=s_lens, [32..34]=softmax weights, [64 ..)=pos_emb[T][128]

    prep_kernel<<<B_SZ, 256, 0, stream>>>(smask, nw, ws);
    posemb_kernel<<<T_LEN, 128, 0, stream>>>(pew1, peb1, pew2, peb2, ws + 64);
    cel_main_kernel<<<B_SZ * (T_LEN / 32), 256, 0, stream>>>(
        emb, smask, tmask, ptw, ptb, lng, lnb, ws, (float*)d_out);
}